// InnerProductDecoder_21869973471660
// MI455X (gfx1250) — compile-verified
//
#include <hip/hip_runtime.h>
#include <math.h>

typedef __attribute__((ext_vector_type(2))) float v2f;
typedef __attribute__((ext_vector_type(4))) float v4f;
typedef __attribute__((ext_vector_type(8))) float v8f;

// ---------------------------------------------------------------------------
// Stage 1: fold the two linear layers.
//   p[k] = dot(w1[k, :],     w2)   k in [0,D)
//   q[k] = dot(w1[D + k, :], w2)
//   c    = dot(b1, w2) + b2
// Tiny (2*D*H MACs); one 256-thread workgroup. Requires D,H <= 256.
// ---------------------------------------------------------------------------
__global__ void __launch_bounds__(256)
fold_weights_kernel(const float* __restrict__ w1, const float* __restrict__ b1,
                    const float* __restrict__ w2, const float* __restrict__ b2,
                    float* __restrict__ p, float* __restrict__ q,
                    float* __restrict__ c, int D, int H) {
  __shared__ float sw2[256];
  __shared__ float red[256];
  const int t = threadIdx.x;
  sw2[t] = (t < H) ? w2[t] : 0.0f;
  __syncthreads();
  if (t < D) {
    float pa = 0.0f, qa = 0.0f;
    const float* rowT = w1 + (size_t)t * H;
    const float* rowB = w1 + (size_t)(D + t) * H;
    for (int h = 0; h < H; ++h) {
      const float wh = sw2[h];
      pa = fmaf(rowT[h], wh, pa);
      qa = fmaf(rowB[h], wh, qa);
    }
    p[t] = pa;
    q[t] = qa;
  }
  red[t] = (t < H) ? b1[t] * sw2[t] : 0.0f;
  __syncthreads();
  for (int s = 128; s > 0; s >>= 1) {
    if (t < s) red[t] += red[t + s];
    __syncthreads();
  }
  if (t == 0) c[0] = red[0] + b2[0];
}

// ---------------------------------------------------------------------------
// Stage 2: [u|v] = z @ [p|q]  via V_WMMA_F32_16X16X4_F32 (exact fp32 MACs).
// One wave (32 lanes) per 16 rows of z. B is the Dx16 matrix whose column 0
// is p, column 1 is q, columns 2..15 zero.
//
// ISA VGPR layouts (CDNA5 ISA):
//   A 16x4 :  lanes 0-15 -> M=lane,    v0=K0, v1=K1
//             lanes16-31 -> M=lane-16, v0=K2, v1=K3
//   B 4x16 :  rows striped across lanes within a VGPR (mirror of A's K split)
//   D 16x16:  VGPR r: lanes 0-15 -> (M=r, N=lane); lanes 16-31 -> (M=r+8, N=lane-16)
// ---------------------------------------------------------------------------
__global__ void __launch_bounds__(32)
gemv_uv_wmma_kernel(const float* __restrict__ z, const float* __restrict__ p,
                    const float* __restrict__ q, float* __restrict__ u,
                    float* __restrict__ v, int D) {
  __shared__ float sp[256];
  __shared__ float sq[256];
  const int lane = threadIdx.x;
  for (int t = lane; t < D; t += 32) { sp[t] = p[t]; sq[t] = q[t]; }
  __syncthreads();

  const int row0  = blockIdx.x * 16;
  const int m     = lane & 15;          // row (A) / column (B,D) index
  const int khalf = (lane >> 4) << 1;   // 0 for lanes 0-15, 2 for lanes 16-31
  const float* zrow = z + (size_t)(row0 + m) * D;

  v8f acc = {};
  for (int kk = 0; kk < D; kk += 4) {
    const int kb = kk + khalf;
    v2f a = *(const v2f*)(zrow + kb);
    v2f b;
    b.x = (m == 0) ? sp[kb]     : (m == 1) ? sq[kb]     : 0.0f;
    b.y = (m == 0) ? sp[kb + 1] : (m == 1) ? sq[kb + 1] : 0.0f;
    acc = __builtin_amdgcn_wmma_f32_16x16x4_f32(
        /*neg_a=*/false, a, /*neg_b=*/false, b,
        /*c_mod=*/(short)0, acc, /*reuse_a=*/false, /*reuse_b=*/false);
  }

  // Column 0 of D = u, column 1 = v.
  if (lane == 0) {
    for (int r = 0; r < 8; ++r) u[row0 + r] = acc[r];
  } else if (lane == 16) {
    for (int r = 0; r < 8; ++r) u[row0 + 8 + r] = acc[r];
  } else if (lane == 1) {
    for (int r = 0; r < 8; ++r) v[row0 + r] = acc[r];
  } else if (lane == 17) {
    for (int r = 0; r < 8; ++r) v[row0 + 8 + r] = acc[r];
  }
}

// ---------------------------------------------------------------------------
// Stage 3: adj[i,j] = 0.5*(sigmoid(u[i]+v[j]+c) + sigmoid(u[j]+v[i]+c)).
// HBM-store-bound: one block per row, each thread writes a float4 (16B
// coalesced stores). sigmoid via __expf -> v_exp_f32.
// ---------------------------------------------------------------------------
__global__ void __launch_bounds__(256)
adj_kernel(const float* __restrict__ u, const float* __restrict__ v,
           const float* __restrict__ c, float* __restrict__ out, int N) {
  const int t = blockIdx.x * blockDim.x + threadIdx.x;
  const int jg_per_row = N >> 2;
  const int i = t / jg_per_row;
  if (i >= N) return;
  const int j0 = (t - i * jg_per_row) << 2;

  const float ui = u[i];
  const float vi = v[i];
  const float cc = c[0];
  const v4f vj = *(const v4f*)(v + j0);
  const v4f uj = *(const v4f*)(u + j0);

  v4f o;
#pragma unroll
  for (int e = 0; e < 4; ++e) {
    const float s_ij = 1.0f / (1.0f + __expf(-(ui + vj[e] + cc)));
    const float s_ji = 1.0f / (1.0f + __expf(-(uj[e] + vi + cc)));
    o[e] = 0.5f * (s_ij + s_ji);
  }
  *(v4f*)(out + (size_t)i * N + j0) = o;
}

// ---------------------------------------------------------------------------
extern "C" void kernel_launch(void* const* d_in, const int* in_sizes, int n_in,
                              void* d_out, int out_size, void* d_ws,
                              size_t ws_size, hipStream_t stream) {
  (void)n_in; (void)out_size; (void)ws_size;

  const float* z  = (const float*)d_in[0];  // [N, D]
  const float* w1 = (const float*)d_in[1];  // [2D, H]
  const float* b1 = (const float*)d_in[2];  // [H]
  const float* w2 = (const float*)d_in[3];  // [H]
  const float* b2 = (const float*)d_in[4];  // [1]
  float* out = (float*)d_out;               // [N, N]

  const int H = in_sizes[3];            // 256
  const int D = in_sizes[1] / (2 * H);  // 256
  const int N = in_sizes[0] / D;        // 1024

  // Workspace: p[D], q[D], u[N], v[N], c[1]  -> ~10.3 KB of d_ws
  float* ws = (float*)d_ws;
  float* p = ws;
  float* q = p + D;
  float* u = q + D;
  float* v = u + N;
  float* c = v + N;

  fold_weights_kernel<<<1, 256, 0, stream>>>(w1, b1, w2, b2, p, q, c, D, H);
  gemv_uv_wmma_kernel<<<N / 16, 32, 0, stream>>>(z, p, q, u, v, D);

  const int total4 = (N * N) / 4;
  adj_kernel<<<(total4 + 255) / 256, 256, 0, stream>>>(u, v, c, out, N);
}